// Net_30571577213456
// MI455X (gfx1250) — compile-verified
//
#include <hip/hip_runtime.h>
#include <hip/hip_bf16.h>
#include <math.h>

// Problem constants (from reference)
constexpr int BPAT   = 400;     // patients
constexpr int NCODES = 8000;    // codes per patient
constexpr int DDIM   = 100;     // code dim
constexpr int SCALER = 4;

constexpr int TILE_ROWS    = 16;                       // rows per wave (WMMA M)
constexpr int TILES_PER_B  = NCODES / TILE_ROWS;       // 500
constexpr int NUM_TILES    = BPAT * TILES_PER_B;       // 200000
constexpr int WAVES_PER_BLOCK = 8;                     // 256 threads
constexpr int NUM_BLOCKS_P1 = NUM_TILES / WAVES_PER_BLOCK; // 25000

typedef float v2f __attribute__((ext_vector_type(2)));
typedef float v8f __attribute__((ext_vector_type(8)));

// ---------------------------------------------------------------------------
// Pass 1: per 16-row tile, compute via V_WMMA_F32_16X16X4_F32:
//   C[16x16] = X_tile[16x100] * Wpanel[100x16]   (cols 0..2 = t, b0, b1)
// then alpha = tanh(t) (output), beta = b1 (output),
// tile partials  s_c = sum_rows tanh(t)*b_c  -> workspace.
// ---------------------------------------------------------------------------
__global__ __launch_bounds__(256) void
pass1_wmma(const float* __restrict__ x,
           const float* __restrict__ dim_weight,   // [D]
           const float* __restrict__ w,            // [(D+4) x 2] row-major
           float* __restrict__ alpha_out,          // [B*N]
           float* __restrict__ beta_out,           // [B*N]
           float* __restrict__ tile_partials)      // [NUM_TILES*2]
{
    __shared__ float ldsA[WAVES_PER_BLOCK][TILE_ROWS * DDIM]; // 8 x 6400B
    __shared__ float ldsB[16 * DDIM];                         // weights [n][k], 6400B
    __shared__ float xfer[WAVES_PER_BLOCK][3][16];            // t/b0/b1 per row

    const int tid  = threadIdx.x;
    const int wave = tid >> 5;
    const int lane = tid & 31;
    const int hi   = lane >> 4;       // 0: lanes 0-15, 1: lanes 16-31
    const int nsel = lane & 15;       // row (A) / column (B,C) index

    // Build transposed weight panel Bp[n][k]: n=0 -> dim_weight,
    // n=1 -> w[:,0], n=2 -> w[:,1], n>=3 -> 0 (padding columns).
    for (int idx = tid; idx < 16 * DDIM; idx += 256) {
        const int n = idx / DDIM;
        const int k = idx - n * DDIM;
        float v = 0.0f;
        if      (n == 0) v = dim_weight[k];
        else if (n == 1) v = w[2 * k + 0];
        else if (n == 2) v = w[2 * k + 1];
        ldsB[idx] = v;
    }

    const int tile = blockIdx.x * WAVES_PER_BLOCK + wave;     // < NUM_TILES
    const float* __restrict__ xb = x + (size_t)tile * (TILE_ROWS * DDIM);

    // Stage 1600 contiguous floats (16 rows x 100) into LDS, coalesced,
    // non-temporal (x is streamed exactly once; don't pollute L2).
    float* __restrict__ la = ldsA[wave];
    #pragma unroll
    for (int j = 0; j < 25; ++j) {
        const int e = (j * 32 + lane) * 2;
        v2f v = __builtin_nontemporal_load((const v2f*)(xb + e));
        *(v2f*)(la + e) = v;
    }
    __syncthreads();

    // K-loop: 25 x V_WMMA_F32_16X16X4_F32.
    // A frag (16x4 f32): VGPR0/1 = K = k0+2*hi, k0+2*hi+1; M = lane&15.
    // B frag (4x16 f32): VGPR0/1 = K = k0+2*hi, k0+2*hi+1; N = lane&15.
    v8f c = {0.f, 0.f, 0.f, 0.f, 0.f, 0.f, 0.f, 0.f};
    const float* __restrict__ arow = la   + nsel * DDIM + 2 * hi;
    const float* __restrict__ brow = ldsB + nsel * DDIM + 2 * hi;
    #pragma unroll
    for (int k0 = 0; k0 < DDIM; k0 += 4) {
        v2f a = *(const v2f*)(arow + k0);
        v2f b = *(const v2f*)(brow + k0);
        c = __builtin_amdgcn_wmma_f32_16x16x4_f32(
                /*neg_a=*/false, a, /*neg_b=*/false, b,
                /*c_mod=*/(short)0, c, /*reuse_a=*/false, /*reuse_b=*/false);
    }

    // C layout: lane holds column N = lane&15; c[v] = C[v + 8*hi][N].
    // Scatter columns 0..2 into the per-wave transfer buffer.
    if (nsel < 3) {
        #pragma unroll
        for (int v = 0; v < 8; ++v)
            xfer[wave][nsel][v + 8 * hi] = c[v];
    }
    // Same-wave DS store->load ordering: wait for outstanding DS ops.
    asm volatile("s_wait_dscnt 0x0" ::: "memory");

    const size_t row_base = (size_t)tile * TILE_ROWS;
    float p0 = 0.0f, p1 = 0.0f;
    if (hi == 0) {
        const int r  = lane;                 // 0..15
        const float t  = xfer[wave][0][r];
        const float b0 = xfer[wave][1][r];
        const float b1 = xfer[wave][2][r];
        const float al = tanhf(t);
        alpha_out[row_base + r] = al;        // coalesced 16-lane store
        p0 = al * b0;
        p1 = al * b1;
    } else {
        const int r = lane - 16;             // 0..15
        beta_out[row_base + r] = xfer[wave][2][r];  // coalesced store
    }

    // Deterministic tree reduction over lanes 0..15 (hi half contributes 0
    // and xor masks <= 8 never cross the half boundary).
    #pragma unroll
    for (int m = 8; m >= 1; m >>= 1) {
        p0 += __shfl_xor(p0, m, 32);
        p1 += __shfl_xor(p1, m, 32);
    }
    if (lane == 0) {
        tile_partials[2 * (size_t)tile + 0] = p0;
        tile_partials[2 * (size_t)tile + 1] = p1;
    }
}

// ---------------------------------------------------------------------------
// Pass 2: one block per patient. Fixed-order reduction of 500 tile partials,
// add x_scaler contribution, write out[b, 0:2].
// ---------------------------------------------------------------------------
__global__ __launch_bounds__(256) void
pass2_reduce(const float* __restrict__ tile_partials,
             const float* __restrict__ x_scaler,   // [B*4]
             const float* __restrict__ w,          // [(D+4) x 2]
             float* __restrict__ out)              // [B*2]
{
    __shared__ float s0[256];
    __shared__ float s1[256];
    const int b   = blockIdx.x;
    const int tid = threadIdx.x;

    float a0 = 0.0f, a1 = 0.0f;
    for (int t = tid; t < TILES_PER_B; t += 256) {
        const size_t idx = 2 * ((size_t)b * TILES_PER_B + t);
        a0 += tile_partials[idx + 0];
        a1 += tile_partials[idx + 1];
    }
    s0[tid] = a0; s1[tid] = a1;
    __syncthreads();
    for (int m = 128; m >= 1; m >>= 1) {
        if (tid < m) { s0[tid] += s0[tid + m]; s1[tid] += s1[tid + m]; }
        __syncthreads();
    }
    if (tid == 0) {
        float o0 = s0[0], o1 = s1[0];
        #pragma unroll
        for (int s = 0; s < SCALER; ++s) {
            const float xs = x_scaler[b * SCALER + s];
            o0 += xs * w[(DDIM + s) * 2 + 0];
            o1 += xs * w[(DDIM + s) * 2 + 1];
        }
        out[2 * b + 0] = o0;
        out[2 * b + 1] = o1;
    }
}

// ---------------------------------------------------------------------------
extern "C" void kernel_launch(void* const* d_in, const int* in_sizes, int n_in,
                              void* d_out, int out_size, void* d_ws, size_t ws_size,
                              hipStream_t stream) {
    const float* x          = (const float*)d_in[0];  // [B,N,D]
    const float* x_scaler   = (const float*)d_in[1];  // [B,4]
    const float* dim_weight = (const float*)d_in[2];  // [D,1]
    const float* w          = (const float*)d_in[3];  // [D+4,2]

    // d_out = out (B*2) ++ alpha (B*N) ++ beta (B*N), all f32
    float* out   = (float*)d_out;
    float* alpha = out + (size_t)BPAT * 2;
    float* beta  = alpha + (size_t)BPAT * NCODES;

    float* tile_partials = (float*)d_ws;   // NUM_TILES*2 floats = 1.6 MB

    pass1_wmma<<<NUM_BLOCKS_P1, 256, 0, stream>>>(x, dim_weight, w,
                                                  alpha, beta, tile_partials);
    pass2_reduce<<<BPAT, 256, 0, stream>>>(tile_partials, x_scaler, w, out);
}